// SE_Module_43310450213261
// MI455X (gfx1250) — compile-verified
//
#include <hip/hip_runtime.h>
#include <math.h>

// SE module: pool(32,512,64,64) -> relu(P.W1^T+b1) -> sigmoid(h.W2^T+b2)
// Phase 1 is HBM-bound (256 MiB @ 23.3 TB/s ~= 11.5us). Phase 2 uses
// V_WMMA_F32_16X16X4_F32 (fp32 matrix pipe) for both tiny GEMMs.

typedef __attribute__((ext_vector_type(2))) float v2f;
typedef __attribute__((ext_vector_type(8))) float v8f;

#define IN_DIM  512
#define OUT_DIM 512
#define MID     32
#define BATCH   32
#define HW      4096   // 64*64

// ---------------- Kernel 1: global average pool ----------------
// grid = BATCH*IN_DIM (16384) blocks, 256 threads. Each block reduces one
// (b,c) row of 4096 floats: 4 x float4 per thread (global_load_b128).
__global__ __launch_bounds__(256) void se_pool_kernel(
    const float* __restrict__ x, float* __restrict__ pooled) {
  const int row = blockIdx.x;                    // b*512 + c
  const float* p = x + (size_t)row * HW;
  const int tid = threadIdx.x;

  float s = 0.f;
#pragma unroll
  for (int i = 0; i < 4; ++i) {
    const float4 v = *(const float4*)(p + i * 1024 + tid * 4);
    s += (v.x + v.y) + (v.z + v.w);
  }
  // wave32 reduction
#pragma unroll
  for (int off = 16; off > 0; off >>= 1) s += __shfl_down(s, off, 32);

  __shared__ float partial[8];
  if ((tid & 31) == 0) partial[tid >> 5] = s;
  __syncthreads();
  if (tid == 0) {
    float t = 0.f;
#pragma unroll
    for (int i = 0; i < 8; ++i) t += partial[i];
    pooled[row] = t * (1.0f / (float)HW);
  }
}

// ---------------- Kernel 2: fused MLP via fp32 WMMA ----------------
// One workgroup, 4 waves (128 threads).
// GEMM1: h[32,32] = relu(P[32,512] x W1^T + b1), 2x2 tiles, one per wave.
// GEMM2: out[32,512] = sigmoid(h x W2^T + b2), 64 tiles round-robined.
__global__ __launch_bounds__(128) void se_mlp_kernel(
    const float* __restrict__ pooled,
    const float* __restrict__ w1, const float* __restrict__ b1,
    const float* __restrict__ w2, const float* __restrict__ b2,
    float* __restrict__ out) {
  __shared__ float h_s[BATCH][MID + 1];   // +1 pad vs bank conflicts

  const int tid  = threadIdx.x;
  const int wid  = tid >> 5;    // wave id 0..3
  const int lane = tid & 31;
  const int half = lane >> 4;   // K-half select (A/B frag layout)
  const int l16  = lane & 15;   // M row (A) / N col (B,C,D)

  // ---- GEMM1: M=batch(32), N=mid(32), K=512 ----
  {
    const int mi = wid >> 1, ni = wid & 1;        // tile coords
    v8f acc = {};
    const float* arow = pooled + (size_t)(mi * 16 + l16) * IN_DIM;
    const float* brow = w1    + (size_t)(ni * 16 + l16) * IN_DIM; // B[k,n]=w1[n,k]
    for (int kb = 0; kb < IN_DIM; kb += 4) {
      const int k = kb + 2 * half;
      v2f a, b;
      a.x = arow[k]; a.y = arow[k + 1];           // contiguous float2
      b.x = brow[k]; b.y = brow[k + 1];
      acc = __builtin_amdgcn_wmma_f32_16x16x4_f32(
          /*neg_a=*/false, a, /*neg_b=*/false, b,
          /*c_mod=*/(short)0, acc, /*reuse_a=*/false, /*reuse_b=*/false);
    }
    const int n = ni * 16 + l16;
    const float bias = b1[n];
#pragma unroll
    for (int j = 0; j < 8; ++j) {                 // D: M = j + 8*half
      const int m = mi * 16 + j + 8 * half;
      const float v = acc[j] + bias;
      h_s[m][n] = v > 0.f ? v : 0.f;              // ReLU
    }
  }
  __syncthreads();

  // ---- GEMM2: M=batch(32), N=out(512), K=32 ----
  for (int t = wid; t < 64; t += 4) {
    const int mi = t & 1;                          // batch tile 0..1
    const int ni = t >> 1;                         // out tile 0..31
    v8f acc = {};
    const int am = mi * 16 + l16;
    const float* brow = w2 + (size_t)(ni * 16 + l16) * MID;  // B[k,n]=w2[n,k]
#pragma unroll
    for (int kb = 0; kb < MID; kb += 4) {
      const int k = kb + 2 * half;
      v2f a, b;
      a.x = h_s[am][k]; a.y = h_s[am][k + 1];
      b.x = brow[k];    b.y = brow[k + 1];
      acc = __builtin_amdgcn_wmma_f32_16x16x4_f32(
          false, a, false, b, (short)0, acc, false, false);
    }
    const int n = ni * 16 + l16;
    const float bias = b2[n];
#pragma unroll
    for (int j = 0; j < 8; ++j) {
      const int m = mi * 16 + j + 8 * half;
      const float v = acc[j] + bias;
      out[m * OUT_DIM + n] = 1.0f / (1.0f + __expf(-v));     // sigmoid
    }
  }
}

extern "C" void kernel_launch(void* const* d_in, const int* in_sizes, int n_in,
                              void* d_out, int out_size, void* d_ws, size_t ws_size,
                              hipStream_t stream) {
  const float* x  = (const float*)d_in[0];   // [32,512,64,64]
  const float* w1 = (const float*)d_in[1];   // [32,512]
  const float* b1 = (const float*)d_in[2];   // [32]
  const float* w2 = (const float*)d_in[3];   // [512,32]
  const float* b2 = (const float*)d_in[4];   // [512]
  float* out    = (float*)d_out;             // [32,512]
  float* pooled = (float*)d_ws;              // 16384 floats = 64 KiB scratch

  se_pool_kernel<<<BATCH * IN_DIM, 256, 0, stream>>>(x, pooled);
  se_mlp_kernel<<<1, 128, 0, stream>>>(pooled, w1, b1, w2, b2, out);
}